// StudentModel_84928683311658
// MI455X (gfx1250) — compile-verified
//
#include <hip/hip_runtime.h>

typedef __attribute__((ext_vector_type(2))) float v2f;
typedef __attribute__((ext_vector_type(8))) float v8f;

#define NEG_SLOPE 0.2f

// ---------------------------------------------------------------------------
// FP32 WMMA GEMM:  C[M,N] = A[M,K] * W[N,K]^T (+ bias[N] if hasBias)
// 128 threads = 4 waves; block tile 64x64, each wave owns a 32x32 tile
// (2x2 of 16x16 V_WMMA_F32_16X16X4_F32).
// K is staged in 64-wide chunks into LDS via gfx1250 async-to-LDS copies
// (global_load_async_to_lds_b128, ASYNCcnt), then fragments are read from
// LDS with a 68-float padded stride -> bank-conflict-free ds_load_b64.
// Fragment layouts follow CDNA5 ISA 7.12.2 (32-bit A: VGPR0={K0,K2},
// VGPR1={K1,K3}; rows on lane&15; C/D: row = r + 8*(lane>=16), col = lane&15).
// ---------------------------------------------------------------------------
__global__ __launch_bounds__(128)
void gemm_wmma_f32(const float* __restrict__ A, const float* __restrict__ W,
                   const float* __restrict__ bias, float* __restrict__ C,
                   int M, int N, int K, int hasBias)
{
    constexpr int KB  = 64;   // K chunk per LDS stage
    constexpr int LDP = 68;   // padded LDS row stride (floats): banks (4r+k)%64
    __shared__ float sA[64 * LDP];
    __shared__ float sB[64 * LDP];

    const int tid  = threadIdx.x;
    const int lane = tid & 31;
    const int wave = tid >> 5;
    const int wm = wave >> 1, wn = wave & 1;
    const int l15 = lane & 15;
    const int hi  = (lane >> 4) & 1;

    const int rowBase = blockIdx.y * 64;
    const int colBase = blockIdx.x * 64;
    const int waveRow = rowBase + wm * 32;
    const int waveCol = colBase + wn * 32;

    v8f acc00 = {}, acc01 = {}, acc10 = {}, acc11 = {};

    // LDS fragment rows for this wave
    const int aR0 = wm * 32 + l15, aR1 = wm * 32 + 16 + l15;
    const int bC0 = wn * 32 + l15, bC1 = wn * 32 + 16 + l15;

    for (int kk = 0; kk < K; kk += KB) {
        // ---- stage 64x64 A and B chunks into LDS (async copy path) ----
        // 1024 b128 segments per tile / 128 threads = 8 per thread per tile
#pragma unroll
        for (int j = 0; j < 8; ++j) {
            int seg = tid + j * 128;
            int row = seg >> 4;                 // 0..63
            int sc  = (seg & 15) * 4;           // float offset within row
            int ga = rowBase + row; if (ga > M - 1) ga = M - 1;  // clamp tails
            int gb = colBase + row; if (gb > N - 1) gb = N - 1;
            unsigned la = (unsigned)(uintptr_t)(&sA[row * LDP + sc]);
            unsigned lb = (unsigned)(uintptr_t)(&sB[row * LDP + sc]);
            const float* pa = A + (size_t)ga * K + kk + sc;
            const float* pb = W + (size_t)gb * K + kk + sc;
            asm volatile("global_load_async_to_lds_b128 %0, %1, off"
                         :: "v"(la), "v"(pa) : "memory");
            asm volatile("global_load_async_to_lds_b128 %0, %1, off"
                         :: "v"(lb), "v"(pb) : "memory");
        }
        asm volatile("s_wait_asynccnt 0x0" ::: "memory");
        __syncthreads();

        // ---- 16 k-steps of fp32 WMMA fed from LDS ----
#pragma unroll
        for (int ks = 0; ks < KB; ks += 4) {
            const int k0 = ks + hi * 2;
            v2f a0 = *(const v2f*)&sA[aR0 * LDP + k0];
            v2f a1 = *(const v2f*)&sA[aR1 * LDP + k0];
            v2f b0 = *(const v2f*)&sB[bC0 * LDP + k0];
            v2f b1 = *(const v2f*)&sB[bC1 * LDP + k0];
            acc00 = __builtin_amdgcn_wmma_f32_16x16x4_f32(false, a0, false, b0, (short)0, acc00, false, false);
            acc01 = __builtin_amdgcn_wmma_f32_16x16x4_f32(false, a0, false, b1, (short)0, acc01, false, false);
            acc10 = __builtin_amdgcn_wmma_f32_16x16x4_f32(false, a1, false, b0, (short)0, acc10, false, false);
            acc11 = __builtin_amdgcn_wmma_f32_16x16x4_f32(false, a1, false, b1, (short)0, acc11, false, false);
        }
        __syncthreads();   // LDS reuse across chunks
    }

    auto store_tile = [&](const v8f& acc, int rb, int cb) {
        int col = cb + l15;
        if (col < N) {
            float bv = hasBias ? bias[col] : 0.0f;
            int row0 = rb + hi * 8;
#pragma unroll
            for (int r = 0; r < 8; ++r) {
                int row = row0 + r;
                if (row < M) C[(size_t)row * N + col] = acc[r] + bv;
            }
        }
    };
    store_tile(acc00, waveRow,      waveCol);
    store_tile(acc01, waveRow,      waveCol + 16);
    store_tile(acc10, waveRow + 16, waveCol);
    store_tile(acc11, waveRow + 16, waveCol + 16);
}

// ---------------------------------------------------------------------------
// attention coefficients: alpha_src/dst[n,h] = dot(h[n,h,:], a[h,:])
// ---------------------------------------------------------------------------
__global__ void attn_coef_kernel(const float* __restrict__ h,
                                 const float* __restrict__ a_s,
                                 const float* __restrict__ a_d,
                                 float* __restrict__ out_s, float* __restrict__ out_d,
                                 int Nn, int H, int C)
{
    int idx = blockIdx.x * blockDim.x + threadIdx.x;
    if (idx >= Nn * H) return;
    int n = idx / H, hh = idx % H;
    const float* hp  = h + (size_t)n * H * C + (size_t)hh * C;
    const float* asp = a_s + (size_t)hh * C;
    const float* adp = a_d + (size_t)hh * C;
    float ss = 0.f, dd = 0.f;
    for (int c = 0; c < C; ++c) { float v = hp[c]; ss += v * asp[c]; dd += v * adp[c]; }
    out_s[idx] = ss; out_d[idx] = dd;
}

__global__ void fill_kernel(float* __restrict__ p, float v, long long n)
{
    long long i = (long long)blockIdx.x * blockDim.x + threadIdx.x;
    if (i < n) p[i] = v;
}

__device__ inline void atomicMaxF(float* addr, float val)
{
    int* ia = (int*)addr;
    int old = __float_as_int(*addr);
    while (__int_as_float(old) < val) {
        int assumed = old;
        old = atomicCAS(ia, assumed, __float_as_int(val));
        if (old == assumed) break;
    }
}

__device__ inline void edge_endpoints(const long long* ei, long long E, long long e,
                                      int& s, int& d)
{
    if (e < E) { s = (int)ei[e]; d = (int)ei[E + e]; }
    else       { s = d = (int)(e - E); }   // appended self-loops
}

// pass 1: e = leaky_relu(as[src]+ad[dst]); segment max over dst
__global__ void edge_lrelu_max(const long long* __restrict__ ei, long long E, long long Etot,
                               const float* __restrict__ as_, const float* __restrict__ ad_,
                               float* __restrict__ ebuf, float* __restrict__ mbuf, int H)
{
    long long idx = (long long)blockIdx.x * blockDim.x + threadIdx.x;
    if (idx >= Etot * H) return;
    long long e = idx / H; int hh = (int)(idx % H);
    int s, d; edge_endpoints(ei, E, e, s, d);
    float v = as_[(size_t)s * H + hh] + ad_[(size_t)d * H + hh];
    v = v > 0.f ? v : NEG_SLOPE * v;
    ebuf[idx] = v;
    atomicMaxF(&mbuf[(size_t)d * H + hh], v);
}

// pass 2: w = exp(e - m[dst]); segment sum over dst
__global__ void edge_expsum(const long long* __restrict__ ei, long long E, long long Etot,
                            const float* __restrict__ mbuf, float* __restrict__ ebuf,
                            float* __restrict__ sbuf, int H)
{
    long long idx = (long long)blockIdx.x * blockDim.x + threadIdx.x;
    if (idx >= Etot * H) return;
    long long e = idx / H; int hh = (int)(idx % H);
    int s, d; edge_endpoints(ei, E, e, s, d);
    float w = __expf(ebuf[idx] - mbuf[(size_t)d * H + hh]);
    ebuf[idx] = w;
    atomicAdd(&sbuf[(size_t)d * H + hh], w);
}

// pass 3: agg[dst,h,:] += h[src,h,:] * (w / (s[dst,h] + eps))
__global__ void edge_scatter(const long long* __restrict__ ei, long long E, long long Etot,
                             const float* __restrict__ hfeat, const float* __restrict__ ebuf,
                             const float* __restrict__ sbuf, float* __restrict__ agg,
                             int H, int C)
{
    long long idx = (long long)blockIdx.x * blockDim.x + threadIdx.x;
    if (idx >= Etot * H) return;
    long long e = idx / H; int hh = (int)(idx % H);
    int s, d; edge_endpoints(ei, E, e, s, d);
    float coef = ebuf[idx] / (sbuf[(size_t)d * H + hh] + 1e-16f);
    const float* hs = hfeat + (size_t)s * H * C + (size_t)hh * C;
    float*       ap = agg   + (size_t)d * H * C + (size_t)hh * C;
    for (int c = 0; c < C; ++c) atomicAdd(&ap[c], hs[c] * coef);
}

// out = elu(agg + bias) + id
__global__ void elu_res_kernel(const float* __restrict__ agg, const float* __restrict__ bias,
                               const float* __restrict__ id, float* __restrict__ out,
                               long long total, int C)
{
    long long i = (long long)blockIdx.x * blockDim.x + threadIdx.x;
    if (i >= total) return;
    int c = (int)(i % C);
    float v = agg[i] + bias[c];
    v = v > 0.f ? v : (__expf(v) - 1.f);
    out[i] = v + id[i];
}

// out[n,c] = mean_h(agg[n,h,c]) + b3[c] + id3[n,c]
__global__ void final_combine(const float* __restrict__ agg, const float* __restrict__ b3,
                              const float* __restrict__ id3, float* __restrict__ out,
                              int Nn, int H, int C)
{
    int i = blockIdx.x * blockDim.x + threadIdx.x;
    if (i >= Nn * C) return;
    int n = i / C, c = i % C;
    float sum = 0.f;
    for (int h = 0; h < H; ++h) sum += agg[(size_t)n * H * C + (size_t)h * C + c];
    out[i] = sum * (1.0f / H) + b3[c] + id3[i];
}

// ---------------------------------------------------------------------------
extern "C" void kernel_launch(void* const* d_in, const int* in_sizes, int n_in,
                              void* d_out, int out_size, void* d_ws, size_t ws_size,
                              hipStream_t stream)
{
    const float*     x   = (const float*)d_in[0];
    const long long* ei  = (const long long*)d_in[1];   // int64 edge_index [2,E]
    const float*     W1  = (const float*)d_in[2];
    const float*     a1s = (const float*)d_in[3];
    const float*     a1d = (const float*)d_in[4];
    const float*     b1  = (const float*)d_in[5];
    const float*     W2  = (const float*)d_in[6];
    const float*     a2s = (const float*)d_in[7];
    const float*     a2d = (const float*)d_in[8];
    const float*     b2  = (const float*)d_in[9];
    const float*     W3  = (const float*)d_in[10];
    const float*     a3s = (const float*)d_in[11];
    const float*     a3d = (const float*)d_in[12];
    const float*     b3  = (const float*)d_in[13];
    const float*     P1W = (const float*)d_in[14];
    const float*     P1b = (const float*)d_in[15];
    const float*     P3W = (const float*)d_in[16];
    const float*     P3b = (const float*)d_in[17];
    float* out = (float*)d_out;

    const int       IN   = 256;
    const int       Nn   = in_sizes[0] / IN;     // 10000
    const long long E    = in_sizes[1] / 2;      // 160000
    const long long Etot = E + Nn;               // with self-loops

    // workspace layout (floats)
    float* ws = (float*)d_ws;
    size_t o = 0;
    float* h_lin = ws + o; o += (size_t)Nn * 320;   // per-layer linear output
    float* agg   = ws + o; o += (size_t)Nn * 320;   // aggregated messages
    float* h_cur = ws + o; o += (size_t)Nn * 256;   // layer activation
    float* idb   = ws + o; o += (size_t)Nn * 256;   // identity projections
    float* asb   = ws + o; o += (size_t)Nn * 16;    // alpha_src
    float* adb   = ws + o; o += (size_t)Nn * 16;    // alpha_dst
    float* mb    = ws + o; o += (size_t)Nn * 16;    // segment max
    float* sb    = ws + o; o += (size_t)Nn * 16;    // segment sum
    float* eb    = ws + o;                          // per-edge coeff, Etot*H

    dim3 blk(128);
    auto gemmGrid = [&](int m, int n) { return dim3((unsigned)((n + 63) / 64), (unsigned)((m + 63) / 64)); };

    auto run_gat = [&](const float* feat_in, const float* Wt, const float* asw, const float* adw, int H) {
        const int C = 32, HC = H * C;
        gemm_wmma_f32<<<gemmGrid(Nn, HC), blk, 0, stream>>>(feat_in, Wt, nullptr, h_lin, Nn, HC, IN, 0);
        long long nh = (long long)Nn * H;
        attn_coef_kernel<<<(unsigned)((nh + 255) / 256), 256, 0, stream>>>(h_lin, asw, adw, asb, adb, Nn, H, C);
        fill_kernel<<<(unsigned)((nh + 255) / 256), 256, 0, stream>>>(mb, -3.0e38f, nh);
        fill_kernel<<<(unsigned)((nh + 255) / 256), 256, 0, stream>>>(sb, 0.f, nh);
        long long na = (long long)Nn * HC;
        fill_kernel<<<(unsigned)((na + 255) / 256), 256, 0, stream>>>(agg, 0.f, na);
        long long ehc = Etot * H;
        unsigned gE = (unsigned)((ehc + 255) / 256);
        edge_lrelu_max<<<gE, 256, 0, stream>>>(ei, E, Etot, asb, adb, eb, mb, H);
        edge_expsum  <<<gE, 256, 0, stream>>>(ei, E, Etot, mb, eb, sb, H);
        edge_scatter <<<gE, 256, 0, stream>>>(ei, E, Etot, h_lin, eb, sb, agg, H, C);
    };

    long long tot256 = (long long)Nn * 256;

    // ---- Layer 1: GAT(256 -> 8x32) + ELU + residual proj ----
    run_gat(x, W1, a1s, a1d, 8);
    gemm_wmma_f32<<<gemmGrid(Nn, 256), blk, 0, stream>>>(x, P1W, P1b, idb, Nn, 256, IN, 1);
    elu_res_kernel<<<(unsigned)((tot256 + 255) / 256), 256, 0, stream>>>(agg, b1, idb, h_cur, tot256, 256);

    // ---- Layer 2: GAT(256 -> 8x32) + ELU + identity residual ----
    run_gat(h_cur, W2, a2s, a2d, 8);
    elu_res_kernel<<<(unsigned)((tot256 + 255) / 256), 256, 0, stream>>>(agg, b2, h_cur, h_cur, tot256, 256);

    // ---- Layer 3: residual proj to 32, GAT(256 -> 10x32, mean) ----
    gemm_wmma_f32<<<gemmGrid(Nn, 32), blk, 0, stream>>>(h_cur, P3W, P3b, idb, Nn, 32, IN, 1);
    run_gat(h_cur, W3, a3s, a3d, 10);
    final_combine<<<(unsigned)(((long long)Nn * 32 + 255) / 256), 256, 0, stream>>>(agg, b3, idb, out, Nn, 10, 32);
}